// CP_model_38104949850431
// MI455X (gfx1250) — compile-verified
//
#include <hip/hip_runtime.h>
#include <math.h>

// Problem dims (fixed by reference)
#define NFRAMES 64
#define PIX     65536            // 256*256
#define NTOT    (NFRAMES * PIX)  // 4,194,304 complex elements
#define SIGMA_F 0.35355339059327373f  // 1/sqrt(8); tau == sigma

typedef __attribute__((ext_vector_type(2))) float v2f;
typedef __attribute__((ext_vector_type(8))) float v8f;

// ---------------------------------------------------------------------------
// complex helpers (float2 = {re, im})
// ---------------------------------------------------------------------------
__device__ __forceinline__ float2 cadd(float2 a, float2 b) { return make_float2(a.x + b.x, a.y + b.y); }
__device__ __forceinline__ float2 csub(float2 a, float2 b) { return make_float2(a.x - b.x, a.y - b.y); }
__device__ __forceinline__ float2 cscale(float2 a, float s) { return make_float2(a.x * s, a.y * s); }

// fp32 WMMA: D(16x16) = A(16x4) * B(4x16) + C   (CDNA5 V_WMMA_F32_16X16X4_F32)
__device__ __forceinline__ v8f wmma4(v2f a, v2f b, v8f c) {
  return __builtin_amdgcn_wmma_f32_16x16x4_f32(false, a, false, b, (short)0, c, false, false);
}

// ---------------------------------------------------------------------------
// Tensor Data Mover (TDM) support: global -> LDS tile DMA (ISA ch. 8).
// This toolchain exposes the 6-arg builtin:
//   (u32x4 g0, i32x8 g1, i32x4 g2, i32x4 g3, i32x8 extra, i32 cpol)
// Guarded: falls back to coalesced VMEM loads if the builtin is absent.
// ---------------------------------------------------------------------------
#if __has_builtin(__builtin_amdgcn_tensor_load_to_lds) && __has_builtin(__builtin_amdgcn_s_wait_tensorcnt)
#define HAVE_TDM 1
#else
#define HAVE_TDM 0
#endif

#if HAVE_TDM
typedef __attribute__((ext_vector_type(4))) unsigned int u32x4;
typedef __attribute__((ext_vector_type(8))) int i32x8;
typedef __attribute__((ext_vector_type(4))) int i32x4;

// Low 32 bits of a generic pointer to LDS == LDS byte offset (ISA 10.2 aperture calc)
__device__ __forceinline__ unsigned lds_byte_off(const void* p) {
  return (unsigned)(uintptr_t)p;
}

// 2D tile load: tile_d0 elements per line, tile_d1 lines, line stride stride0
// (all in data_size=8B units). Optional LDS padding after each pad-interval.
__device__ __forceinline__ void tdm_load_2d(const void* gptr, unsigned lds_off,
                                            unsigned tile_d0, unsigned tile_d1,
                                            unsigned tens_d0, unsigned tens_d1,
                                            unsigned stride0,
                                            unsigned pad_en, unsigned pad_interval_code,
                                            unsigned pad_amount_code) {
  unsigned long long ga = (unsigned long long)(uintptr_t)gptr;
  u32x4 g0;
  g0.x = 1u;                                         // count=1 (valid user descriptor)
  g0.y = lds_off;                                    // lds_addr (bytes)
  g0.z = (unsigned)(ga & 0xFFFFFFFFu);               // global_addr[31:0]
  g0.w = (unsigned)((ga >> 32) & 0x01FFFFFFu) | (2u << 30);  // global_addr[56:32] | type=2
  // group1: data_size=3 (8B), pad controls, dims/strides (see ISA 8.4 bit layout)
  unsigned w0 = (3u << 16) | (pad_en << 20) | (pad_interval_code << 22) | (pad_amount_code << 25);
  i32x8 g1;
  g1[0] = (int)w0;
  g1[1] = (int)((tens_d0 & 0xFFFFu) << 16);                          // tensor_dim0[15:0]
  g1[2] = (int)((tens_d0 >> 16) | ((tens_d1 & 0xFFFFu) << 16));      // td0[31:16] | td1[15:0]
  g1[3] = (int)((tens_d1 >> 16) | ((tile_d0 & 0xFFFFu) << 16));      // td1[31:16] | tile_dim0
  g1[4] = (int)(tile_d1 & 0xFFFFu);                                  // tile_dim1 (tile_dim2=0)
  g1[5] = (int)stride0;                                              // tensor_dim0_stride[31:0]
  g1[6] = 0;                                                         // stride0 hi / stride1 lo
  g1[7] = 0;
  i32x4 z4 = {0, 0, 0, 0};
  i32x8 z8 = {0, 0, 0, 0, 0, 0, 0, 0};
  __builtin_amdgcn_tensor_load_to_lds(g0, g1, z4, z4, z8, 0);
}

__device__ __forceinline__ void tdm_wait0() { __builtin_amdgcn_s_wait_tensorcnt(0); }
#endif

// ---------------------------------------------------------------------------
// norm reduction: acc += sum(dr^2 + di^2)
// ---------------------------------------------------------------------------
__global__ void reduce_norm2_kernel(const float* __restrict__ dr,
                                    const float* __restrict__ di,
                                    float* __restrict__ acc) {
  __shared__ float sm[256];
  float s = 0.f;
  for (int i = blockIdx.x * blockDim.x + threadIdx.x; i < NTOT; i += gridDim.x * blockDim.x) {
    float a = dr[i], b = di[i];
    s += a * a + b * b;
  }
  sm[threadIdx.x] = s;
  __syncthreads();
  for (int off = 128; off > 0; off >>= 1) {
    if ((int)threadIdx.x < off) sm[threadIdx.x] += sm[threadIdx.x + off];
    __syncthreads();
  }
  if (threadIdx.x == 0) atomicAdd(acc, sm[0]);
}

__global__ void pack_scale_kernel(const float* __restrict__ dr,
                                  const float* __restrict__ di,
                                  const float* __restrict__ norm2,
                                  float2* __restrict__ D, float2* __restrict__ T) {
  int i = blockIdx.x * blockDim.x + threadIdx.x;
  if (i >= NTOT) return;
  float rn = rsqrtf(norm2[0]);
  float2 v = make_float2(dr[i] * rn, di[i] * rn);
  D[i] = v;
  T[i] = v;
}

// ---------------------------------------------------------------------------
// Row-direction 256-pt radix-2 DIT FFT; one block (128 thr) per contiguous line.
// Line staged into LDS by the Tensor Data Mover when available.
// sign = -1 forward, +1 inverse; scales by 1/16 (ortho, per 1D pass).
// ---------------------------------------------------------------------------
__global__ void fft_rows_kernel(float2* __restrict__ buf, float sign) {
  __shared__ float2 S[256];
#if HAVE_TDM
  __shared__ float2 Sstage[256];
#endif
  int base = blockIdx.x * 256;
  int t = threadIdx.x;  // 128 threads

#if HAVE_TDM
  if (t < 32) {  // wave 0 issues the DMA: 1D tile of 256 x 8B
    tdm_load_2d(buf + base, lds_byte_off(Sstage), 256u, 0u, 256u, 1u, 256u, 0u, 0u, 0u);
    tdm_wait0();
  }
  __syncthreads();
  for (int e = t; e < 256; e += 128) {
    int r = (int)(__brev((unsigned)e) >> 24);
    S[r] = Sstage[e];
  }
#else
  for (int e = t; e < 256; e += 128) {
    int r = (int)(__brev((unsigned)e) >> 24);
    S[r] = buf[base + e];
  }
#endif
  __syncthreads();

  for (int s = 1; s <= 8; ++s) {
    int half = 1 << (s - 1);
    int grp = t >> (s - 1);
    int k = t & (half - 1);
    int i0 = (grp << s) + k;
    int i1 = i0 + half;
    float ang = sign * 3.14159265358979323846f * (float)k / (float)half;
    float sw, cw;
    __sincosf(ang, &sw, &cw);
    float2 a = S[i0], b = S[i1];
    float2 bw = make_float2(b.x * cw - b.y * sw, b.x * sw + b.y * cw);
    S[i0] = make_float2(a.x + bw.x, a.y + bw.y);
    S[i1] = make_float2(a.x - bw.x, a.y - bw.y);
    __syncthreads();
  }

  for (int e = t; e < 256; e += 128) {
    buf[base + e] = cscale(S[e], 0.0625f);
  }
}

// ---------------------------------------------------------------------------
// Column-direction FFT: one block (256 thr) handles a 256x16 tile (16 columns)
// of one frame. Tile is DMA'd into LDS by TDM with hardware row padding
// (pad_interval code 4 = 32 DWORDs, pad_amount code 1 = 2 DWORDs -> row
// stride 17 float2) to spread LDS banks; falls back to coalesced loads.
// ---------------------------------------------------------------------------
#define CSTRIDE 17  // padded float2 stride per row in LDS

__global__ void fft_cols_kernel(float2* __restrict__ buf, float sign) {
  __shared__ float2 S[256 * CSTRIDE];  // 34,816 B
  int frame = blockIdx.x >> 4;
  int c0 = (blockIdx.x & 15) * 16;
  int t = threadIdx.x;  // 256 threads
  float2* gbase = buf + frame * PIX + c0;

#if HAVE_TDM
  if (t < 32) {  // 2D tile: 16 elems/line, 256 lines, global line stride 256 elems
    tdm_load_2d(gbase, lds_byte_off(S), 16u, 256u, 16u, 256u, 256u,
                /*pad_en=*/1u, /*interval 32 DW=*/4u, /*amount 2 DW=*/1u);
    tdm_wait0();
  }
  __syncthreads();
#else
  {
    int r = t;
    const float2* rp = gbase + r * 256;
    for (int c = 0; c < 16; ++c) S[r * CSTRIDE + c] = rp[c];
  }
  __syncthreads();
#endif

  // in-place bit-reversal permutation of the 256 rows (16 cols each)
  {
    int r = t;
    int br = (int)(__brev((unsigned)r) >> 24);
    if (r < br) {
      for (int c = 0; c < 16; ++c) {
        float2 tmp = S[r * CSTRIDE + c];
        S[r * CSTRIDE + c] = S[br * CSTRIDE + c];
        S[br * CSTRIDE + c] = tmp;
      }
    }
  }
  __syncthreads();

  int col = t & 15;
  int j0 = t >> 4;  // 0..15; each thread does 8 butterflies/stage/column
  for (int s = 1; s <= 8; ++s) {
    int half = 1 << (s - 1);
    for (int it = 0; it < 8; ++it) {
      int j = j0 + (it << 4);  // 0..127
      int grp = j >> (s - 1);
      int k = j & (half - 1);
      int i0 = (grp << s) + k;
      int i1 = i0 + half;
      float ang = sign * 3.14159265358979323846f * (float)k / (float)half;
      float sw, cw;
      __sincosf(ang, &sw, &cw);
      float2 a = S[i0 * CSTRIDE + col], b = S[i1 * CSTRIDE + col];
      float2 bw = make_float2(b.x * cw - b.y * sw, b.x * sw + b.y * cw);
      S[i0 * CSTRIDE + col] = make_float2(a.x + bw.x, a.y + bw.y);
      S[i1 * CSTRIDE + col] = make_float2(a.x - bw.x, a.y - bw.y);
    }
    __syncthreads();
  }

  {
    int r = t;
    float2* rp = gbase + r * 256;
    for (int c = 0; c < 16; ++c) rp[c] = cscale(S[r * CSTRIDE + c], 0.0625f);
  }
}

// ---------------------------------------------------------------------------
// state init: x1=x2=u1=u2=startpoint, y1=y2=0
// ---------------------------------------------------------------------------
__global__ void init_state_kernel(const float2* __restrict__ SP,
                                  float2* x1, float2* x2, float2* u1, float2* u2,
                                  float2* y1, float2* y2) {
  int i = blockIdx.x * blockDim.x + threadIdx.x;
  if (i >= NTOT) return;
  float2 v = SP[i];
  x1[i] = v; x2[i] = v; u1[i] = v; u2[i] = v;
  float2 z = make_float2(0.f, 0.f);
  y1[i] = z; y2[i] = z;
}

__global__ void add_u_kernel(const float2* u1, const float2* u2, float2* T) {
  int i = blockIdx.x * blockDim.x + threadIdx.x;
  if (i >= NTOT) return;
  T[i] = cadd(u1[i], u2[i]);
}

// y1 = argf1 - sigma*(argf1 + d)/(1+sigma),  argf1 = y1 + sigma*Ku1 (Ku1 in T)
__global__ void y1_update_kernel(float2* y1, const float2* T, const float2* D) {
  int i = blockIdx.x * blockDim.x + threadIdx.x;
  if (i >= NTOT) return;
  float2 arg = cadd(y1[i], cscale(T[i], SIGMA_F));
  float kk = SIGMA_F / (1.f + SIGMA_F);
  y1[i] = csub(arg, cscale(cadd(arg, D[i]), kk));
}

// y2 = argf2 - soft(argf2, lamS), argf2 = y2 + sigma*Ku2, Ku2 = temporal diff(u2)
__global__ void y2_update_kernel(float2* y2, const float2* u2,
                                 const float* __restrict__ lambdaS, int layer) {
  int i = blockIdx.x * blockDim.x + threadIdx.x;
  if (i >= NTOT) return;
  int f = i >> 16;
  float2 ku = make_float2(0.f, 0.f);
  if (f < NFRAMES - 1) ku = csub(u2[i + PIX], u2[i]);
  float2 arg = cadd(y2[i], cscale(ku, SIGMA_F));
  float lamS = fmaxf(lambdaS[layer], 0.f);
  float mag = sqrtf(arg.x * arg.x + arg.y * arg.y);
  float2 soft = make_float2(0.f, 0.f);
  if (mag > lamS && mag > 1e-20f) soft = cscale(arg, (mag - lamS) / mag);
  y2[i] = csub(arg, soft);
}

// Kadjy1 in Ka1 (=ifft2(y1)); builds argg1 -> A1, updates x2/u2 with argg2
__global__ void adj_update_kernel(const float2* __restrict__ Ka1,
                                  const float2* __restrict__ y2,
                                  const float2* __restrict__ x1,
                                  float2* x2, float2* A1, float2* u2) {
  int i = blockIdx.x * blockDim.x + threadIdx.x;
  if (i >= NTOT) return;
  int f = i >> 16, p = i & 0xFFFF;
  float2 k1 = Ka1[i];
  float2 k2;
  if (f == 0)                k2 = csub(k1, y2[i]);
  else if (f == NFRAMES - 1) k2 = cadd(k1, y2[(NFRAMES - 2) * PIX + p]);
  else                       k2 = csub(k1, csub(y2[i], y2[i - PIX]));
  A1[i] = csub(x1[i], cscale(k1, SIGMA_F));     // argg1 (tau == sigma)
  float2 xo = x2[i];
  float2 ag2 = csub(xo, cscale(k2, SIGMA_F));   // argg2
  x2[i] = ag2;
  u2[i] = csub(cscale(ag2, 2.f), xo);           // over-relaxation theta=1
}

// ---------------------------------------------------------------------------
// Gram: G(64x64, complex) = M^H M, M[p][f] = A1[f*PIX + p]
// grid = 16 tiles * 64 K-slabs, blockDim = 32 (one wave). fp32 WMMA, atomics.
// ---------------------------------------------------------------------------
__global__ void gram_wmma_kernel(const float2* __restrict__ M, float* __restrict__ G) {
  int tile = blockIdx.x & 15;
  int slab = blockIdx.x >> 4;
  int ti = (tile >> 2) * 16;     // G row block  (frame i)
  int tj = (tile & 3) * 16;      // G col block  (frame j)
  int p0 = slab * 1024;          // 65536 / 64 slabs
  int lane = threadIdx.x;
  int mrow = lane & 15;
  int khalf = (lane >> 4) << 1;  // 0 or 2

  v8f acc_re = {}; v8f acc_im = {};
  const float2* rowi = M + (ti + mrow) * PIX;
  const float2* rowj = M + (tj + mrow) * PIX;
  for (int kc = 0; kc < 1024; kc += 4) {
    int p = p0 + kc + khalf;
    __builtin_prefetch((const void*)(rowi + p + 64), 0, 1);
    float2 a0 = rowi[p], a1 = rowi[p + 1];
    float2 b0 = rowj[p], b1 = rowj[p + 1];
    v2f Ar  = {a0.x, a1.x};
    v2f Ai  = {a0.y, a1.y};
    v2f nAi = {-a0.y, -a1.y};
    v2f Br  = {b0.x, b1.x};
    v2f Bi  = {b0.y, b1.y};
    // G = conj(M_i)^T M_j :  re = ArBr + AiBi ; im = ArBi - AiBr
    acc_re = wmma4(Ar, Br, acc_re);
    acc_re = wmma4(Ai, Bi, acc_re);
    acc_im = wmma4(Ar, Bi, acc_im);
    acc_im = wmma4(nAi, Br, acc_im);
  }
  int n = lane & 15;
  int mbase = (lane >> 4) << 3;
  for (int vg = 0; vg < 8; ++vg) {
    int gi = ti + vg + mbase, gj = tj + n;
    atomicAdd(&G[(gi * 64 + gj) * 2 + 0], acc_re[vg]);
    atomicAdd(&G[(gi * 64 + gj) * 2 + 1], acc_im[vg]);
  }
}

// ---------------------------------------------------------------------------
// One-block Hermitian Jacobi eigensolver for G (64x64) + low-rank prox weight:
// W = V diag(soft(S)/S) V^H,  S = sqrt(max(eig,0)), thr = Smax * tau * relu(lamL)
// ---------------------------------------------------------------------------
__global__ void jacobi_svd_kernel(const float2* __restrict__ Gin, float2* __restrict__ W,
                                  const float* __restrict__ lambdaL, int layer) {
  extern __shared__ float2 jsm[];
  float2* G = jsm;          // 64*64
  float2* V = jsm + 4096;   // 64*64
  __shared__ int perm[64];
  __shared__ int pA[32], qA[32];
  __shared__ float cA[32];
  __shared__ float2 sA[32];
  __shared__ float svals[64], ratio[64], smax_sh;
  int tid = threadIdx.x;  // 256 threads

  for (int idx = tid; idx < 4096; idx += blockDim.x) {
    G[idx] = Gin[idx];
    int r = idx >> 6, c = idx & 63;
    V[idx] = make_float2(r == c ? 1.f : 0.f, 0.f);
  }
  if (tid < 64) perm[tid] = tid;
  __syncthreads();

  for (int sweep = 0; sweep < 12; ++sweep) {
    for (int round = 0; round < 63; ++round) {
      if (tid < 32) {
        int p = perm[tid], q = perm[63 - tid];
        if (p > q) { int t = p; p = q; q = t; }
        pA[tid] = p; qA[tid] = q;
        float2 apq = G[p * 64 + q];
        float app = G[p * 64 + p].x, aqq = G[q * 64 + q].x;
        float mag = sqrtf(apq.x * apq.x + apq.y * apq.y);
        float cc = 1.f;
        float2 ss = make_float2(0.f, 0.f);
        if (mag > 1e-12f) {
          float tt = (aqq - app) / (2.f * mag);
          float tn = copysignf(1.f, tt) / (fabsf(tt) + sqrtf(1.f + tt * tt));
          cc = rsqrtf(1.f + tn * tn);
          float sr = tn * cc;
          ss = make_float2(sr * (apq.x / mag), sr * (apq.y / mag));
        }
        cA[tid] = cc; sA[tid] = ss;
      }
      __syncthreads();
      // right multiply: X <- X R  (col p' = c*colp - conj(s)*colq ; col q' = s*colp + c*colq)
      for (int w = tid; w < 64 * 32; w += blockDim.x) {
        int pair = w & 31, i = w >> 5;
        int p = pA[pair], q = qA[pair];
        float cc = cA[pair]; float2 s = sA[pair];
        float2 gp = G[i * 64 + p], gq = G[i * 64 + q];
        float2 np = make_float2(cc * gp.x - (s.x * gq.x + s.y * gq.y),
                                cc * gp.y - (s.x * gq.y - s.y * gq.x));
        float2 nq = make_float2(s.x * gp.x - s.y * gp.y + cc * gq.x,
                                s.x * gp.y + s.y * gp.x + cc * gq.y);
        G[i * 64 + p] = np; G[i * 64 + q] = nq;
        float2 vp = V[i * 64 + p], vq = V[i * 64 + q];
        float2 wp = make_float2(cc * vp.x - (s.x * vq.x + s.y * vq.y),
                                cc * vp.y - (s.x * vq.y - s.y * vq.x));
        float2 wq = make_float2(s.x * vp.x - s.y * vp.y + cc * vq.x,
                                s.x * vp.y + s.y * vp.x + cc * vq.y);
        V[i * 64 + p] = wp; V[i * 64 + q] = wq;
      }
      __syncthreads();
      // left multiply: G <- R^H G (row p' = c*rowp - s*rowq ; row q' = conj(s)*rowp + c*rowq)
      for (int w = tid; w < 64 * 32; w += blockDim.x) {
        int pair = w & 31, j = w >> 5;
        int p = pA[pair], q = qA[pair];
        float cc = cA[pair]; float2 s = sA[pair];
        float2 gp = G[p * 64 + j], gq = G[q * 64 + j];
        float2 np = make_float2(cc * gp.x - (s.x * gq.x - s.y * gq.y),
                                cc * gp.y - (s.x * gq.y + s.y * gq.x));
        float2 nq = make_float2((s.x * gp.x + s.y * gp.y) + cc * gq.x,
                                (s.x * gp.y - s.y * gp.x) + cc * gq.y);
        G[p * 64 + j] = np; G[q * 64 + j] = nq;
      }
      __syncthreads();
      if (tid == 0) {  // rotate players 1..63 (player 0 fixed)
        int last = perm[63];
        for (int k = 63; k > 1; --k) perm[k] = perm[k - 1];
        perm[1] = last;
      }
      __syncthreads();
    }
  }

  if (tid < 64) svals[tid] = sqrtf(fmaxf(G[tid * 64 + tid].x, 0.f));
  __syncthreads();
  if (tid == 0) {
    float m = 0.f;
    for (int k = 0; k < 64; ++k) m = fmaxf(m, svals[k]);
    smax_sh = m;
  }
  __syncthreads();
  float lamL = fmaxf(lambdaL[layer], 0.f);
  float thr = smax_sh * SIGMA_F * lamL;  // tau == sigma
  if (tid < 64) {
    float s = svals[tid];
    ratio[tid] = (s > 1e-20f) ? fmaxf(s - thr, 0.f) / s : 0.f;
  }
  __syncthreads();
  // W[i][j] = sum_k V[i][k] * ratio[k] * conj(V[j][k])
  for (int idx = tid; idx < 4096; idx += blockDim.x) {
    int i = idx >> 6, j = idx & 63;
    float2 acc = make_float2(0.f, 0.f);
    for (int k = 0; k < 64; ++k) {
      float2 a = V[i * 64 + k];
      float2 b = V[j * 64 + k];
      float r = ratio[k];
      acc.x += r * (a.x * b.x + a.y * b.y);
      acc.y += r * (a.y * b.x - a.x * b.y);
    }
    W[idx] = acc;
  }
}

// ---------------------------------------------------------------------------
// x1_new = M W (65536x64 * 64x64 complex), u1 = 2*x1_new - x1_old.
// One wave per 16x16 output tile; fp32 WMMA, K = 64 in chunks of 4.
// ---------------------------------------------------------------------------
__global__ void mw_wmma_kernel(const float2* __restrict__ M, const float2* __restrict__ W,
                               float2* __restrict__ x1, float2* __restrict__ u1) {
  int pt = blockIdx.x >> 2;   // 4096 pixel tiles
  int ft = blockIdx.x & 3;    // 4 frame tiles
  int p0 = pt * 16;
  int f0 = ft * 16;
  int lane = threadIdx.x;
  int mrow = lane & 15;           // pixel row of A tile / col of B tile
  int khalf = (lane >> 4) << 1;   // 0 or 2

  v8f acc_re = {}; v8f acc_im = {};
  for (int k = 0; k < NFRAMES; k += 4) {
    int kk = k + khalf;
    float2 a0 = M[kk * PIX + p0 + mrow];
    float2 a1 = M[(kk + 1) * PIX + p0 + mrow];
    float2 b0 = W[kk * 64 + f0 + mrow];
    float2 b1 = W[(kk + 1) * 64 + f0 + mrow];
    v2f Ar  = {a0.x, a1.x};
    v2f Ai  = {a0.y, a1.y};
    v2f nAi = {-a0.y, -a1.y};
    v2f Br  = {b0.x, b1.x};
    v2f Bi  = {b0.y, b1.y};
    // complex product: re = ArBr - AiBi ; im = ArBi + AiBr
    acc_re = wmma4(Ar, Br, acc_re);
    acc_re = wmma4(nAi, Bi, acc_re);
    acc_im = wmma4(Ar, Bi, acc_im);
    acc_im = wmma4(Ai, Br, acc_im);
  }
  int n = lane & 15;
  int mbase = (lane >> 4) << 3;
  for (int vg = 0; vg < 8; ++vg) {
    int idx = (f0 + n) * PIX + p0 + vg + mbase;
    float2 xo = x1[idx];
    float2 xn = make_float2(acc_re[vg], acc_im[vg]);
    x1[idx] = xn;
    u1[idx] = csub(cscale(xn, 2.f), xo);
  }
}

__global__ void out_sum_kernel(const float2* x1, const float2* x2, float2* out) {
  int i = blockIdx.x * blockDim.x + threadIdx.x;
  if (i >= NTOT) return;
  out[i] = cadd(x1[i], x2[i]);
}

// ---------------------------------------------------------------------------
// host launcher
// ---------------------------------------------------------------------------
extern "C" void kernel_launch(void* const* d_in, const int* in_sizes, int n_in,
                              void* d_out, int out_size, void* d_ws, size_t ws_size,
                              hipStream_t stream) {
  (void)in_sizes; (void)n_in; (void)out_size; (void)ws_size;
  const float* dr   = (const float*)d_in[0];
  const float* di   = (const float*)d_in[1];
  const float* lamS = (const float*)d_in[2];
  const float* lamL = (const float*)d_in[3];

  char* ws = (char*)d_ws;
  const size_t CSZ = (size_t)NTOT * sizeof(float2);
  float2* D  = (float2*)(ws + 0 * CSZ);
  float2* T  = (float2*)(ws + 1 * CSZ);
  float2* y1 = (float2*)(ws + 2 * CSZ);
  float2* y2 = (float2*)(ws + 3 * CSZ);
  float2* x1 = (float2*)(ws + 4 * CSZ);
  float2* x2 = (float2*)(ws + 5 * CSZ);
  float2* u1 = (float2*)(ws + 6 * CSZ);
  float2* u2 = (float2*)(ws + 7 * CSZ);
  float2* A1 = (float2*)(ws + 8 * CSZ);
  float*  nrm = (float*)(ws + 9 * CSZ);
  float*  Gm  = (float*)(ws + 9 * CSZ + 256);
  float2* Wm  = (float2*)(ws + 9 * CSZ + 256 + 64 * 64 * 2 * sizeof(float));

  const int gElem = (NTOT + 255) / 256;  // 16384
  const int gRows = NFRAMES * 256;       // 16384 row lines
  const int gCols = NFRAMES * 16;        // 1024 column tiles (16 cols each)

  (void)hipMemsetAsync(nrm, 0, sizeof(float), stream);
  reduce_norm2_kernel<<<2048, 256, 0, stream>>>(dr, di, nrm);
  pack_scale_kernel<<<gElem, 256, 0, stream>>>(dr, di, nrm, D, T);
  // startpoint = ifft2o(d)
  fft_rows_kernel<<<gRows, 128, 0, stream>>>(T, +1.f);
  fft_cols_kernel<<<gCols, 256, 0, stream>>>(T, +1.f);
  init_state_kernel<<<gElem, 256, 0, stream>>>(T, x1, x2, u1, u2, y1, y2);

  for (int layer = 0; layer < 8; ++layer) {
    add_u_kernel<<<gElem, 256, 0, stream>>>(u1, u2, T);
    fft_rows_kernel<<<gRows, 128, 0, stream>>>(T, -1.f);   // fft2o(u1+u2)
    fft_cols_kernel<<<gCols, 256, 0, stream>>>(T, -1.f);
    y1_update_kernel<<<gElem, 256, 0, stream>>>(y1, T, D);
    y2_update_kernel<<<gElem, 256, 0, stream>>>(y2, u2, lamS, layer);
    (void)hipMemcpyAsync(T, y1, CSZ, hipMemcpyDeviceToDevice, stream);
    fft_rows_kernel<<<gRows, 128, 0, stream>>>(T, +1.f);   // Kadjy1 = ifft2o(y1)
    fft_cols_kernel<<<gCols, 256, 0, stream>>>(T, +1.f);
    adj_update_kernel<<<gElem, 256, 0, stream>>>(T, y2, x1, x2, A1, u2);
    (void)hipMemsetAsync(Gm, 0, 64 * 64 * 2 * sizeof(float), stream);
    gram_wmma_kernel<<<16 * 64, 32, 0, stream>>>(A1, Gm);
    jacobi_svd_kernel<<<1, 256, 2 * 4096 * sizeof(float2), stream>>>(
        (const float2*)Gm, Wm, lamL, layer);
    mw_wmma_kernel<<<(PIX / 16) * (NFRAMES / 16), 32, 0, stream>>>(A1, Wm, x1, u1);
  }

  out_sum_kernel<<<gElem, 256, 0, stream>>>(x1, x2, (float2*)d_out);
}